// HyperNet_9088150798945
// MI455X (gfx1250) — compile-verified
//
#include <hip/hip_runtime.h>
#include <math.h>

typedef __attribute__((ext_vector_type(2))) float v2f;
typedef __attribute__((ext_vector_type(8))) float v8f;

constexpr int   DIM      = 64;   // full Lorentz dim
constexpr int   DM       = 63;   // spatial dim
constexpr float MAXNORM  = 2.5f;
constexpr float MARGIN_C = 8.0f;
constexpr float EPS_C    = 1e-6f;

__device__ __forceinline__ float gelu_exact(float x) {
    return 0.5f * x * (1.0f + erff(x * 0.70710678118654752f));
}

// ---------------------------------------------------------------------------
// Kernel A: per-relation precompute. grid = (NR, 2) [y: 0=head,1=tail], 64 thr
//   Rot[r]  : 64x64 embedded rotation (row0/col0 = e0 passthrough),
//             built as a product of 63 Householder reflections via rank-1
//             updates (identical math to the reference's matmul chain).
//   off[r]  : _do() offset = expmap(center, 0.1 * unit-tangent)
//   y2[r]   : ||off||^2 (reused by mobius_add)
// ---------------------------------------------------------------------------
__global__ __launch_bounds__(64)
void rel_precompute_kernel(const float* __restrict__ W_h,
                           const float* __restrict__ W_t,
                           const float* __restrict__ cen_h,
                           const float* __restrict__ dir_h,
                           const float* __restrict__ cen_t,
                           const float* __restrict__ dir_t,
                           const float* __restrict__ flip_h,
                           const float* __restrict__ flip_t,
                           float* __restrict__ RotH, float* __restrict__ RotT,
                           float* __restrict__ offH, float* __restrict__ offT,
                           float* __restrict__ y2H,  float* __restrict__ y2T)
{
    const int r    = blockIdx.x;
    const int side = blockIdx.y;        // 0 = head, 1 = tail
    const int tid  = threadIdx.x;

    const float* Wsrc = (side ? W_t : W_h) + (size_t)r * (DM * DM);
    const float  flip = side ? flip_t[0] : flip_h[0];
    float*       Rot  = (side ? RotT : RotH) + (size_t)r * (DIM * DIM);

    __shared__ float Wl[DM * 64];       // gelu(W), row stride 64
    __shared__ float rs[DM];            // per-row squared norms
    __shared__ float prod[DM * 65];     // running product, stride 65 (no bank conflicts)

    for (int idx = tid; idx < DM * DM; idx += 64) {
        int i = idx / DM, k = idx - i * DM;
        Wl[i * 64 + k] = gelu_exact(Wsrc[idx]);
    }
    __syncthreads();
    if (tid < DM) {
        float s = 0.f;
        for (int k = 0; k < DM; ++k) { float w = Wl[tid * 64 + k]; s += w * w; }
        rs[tid] = s;
        for (int k = 0; k < DM; ++k) prod[tid * 65 + k] = (tid == k) ? 1.f : 0.f;
    }
    __syncthreads();
    // prod <- prod * (I - 2 w_i w_i^T / ||w_i||^2), i = 0..62.
    // Row tid of prod depends only on row tid: no barriers needed in the chain.
    if (tid < DM) {
        for (int i = 0; i < DM; ++i) {
            float v = 0.f;
            for (int k = 0; k < DM; ++k) v += prod[tid * 65 + k] * Wl[i * 64 + k];
            float c = 2.f * v / rs[i];
            for (int l = 0; l < DM; ++l) prod[tid * 65 + l] -= c * Wl[i * 64 + l];
        }
    }
    __syncthreads();
    // Embed into 64x64 (time component passthrough), flip last column.
    for (int idx = tid; idx < DIM * DIM; idx += 64) {
        int i = idx >> 6, j = idx & 63;
        float val;
        if (i == 0 || j == 0) val = (i == 0 && j == 0) ? 1.f : 0.f;
        else {
            val = prod[(i - 1) * 65 + (j - 1)];
            if (j == DIM - 1) val *= flip;
        }
        Rot[idx] = val;
    }
    // Offset vector (trivial; one lane)
    if (tid == 0) {
        const float* c  = (side ? cen_t : cen_h) + r * DIM;
        const float* dd = (side ? dir_t : dir_h) + r * DIM;
        float inner = 0.f;
        for (int k = 0; k < DIM; ++k) inner += c[k] * dd[k];
        float tvec[DIM]; float tn = 0.f;
        for (int k = 0; k < DIM; ++k) { float tv = dd[k] + inner * c[k]; tvec[k] = tv; tn += tv * tv; }
        tn = sqrtf(tn) + EPS_C;
        float l2 = 0.f;
        for (int k = 0; k < DIM; ++k) { tvec[k] = 0.1f * tvec[k] / tn; l2 += tvec[k] * tvec[k]; }
        l2 -= 2.f * tvec[0] * tvec[0];              // Lorentz inner <u,u>
        l2 = fmaxf(l2, 1e-8f);
        float nrm = fminf(sqrtf(l2), MAXNORM);
        float ch = coshf(nrm), sh = sinhf(nrm) / nrm;
        float* off = (side ? offT : offH) + r * DIM;
        float y2 = 0.f;
        for (int k = 0; k < DIM; ++k) { float o = ch * c[k] + sh * tvec[k]; off[k] = o; y2 += o * o; }
        if (side) y2T[r] = y2; else y2H[r] = y2;
    }
}

// ---------------------------------------------------------------------------
// Kernel B: transformed head per batch row. grid = B, 64 threads.
//   TH[b] = mobius_add( flg(ent[u_b]) , offH[r_b] )
// ---------------------------------------------------------------------------
__global__ __launch_bounds__(64)
void head_kernel(const float* __restrict__ ent, const int* __restrict__ u_idx,
                 const int* __restrict__ r_idx,
                 const float* __restrict__ RotH, const float* __restrict__ offH,
                 const float* __restrict__ y2H, float* __restrict__ TH)
{
    const int b = blockIdx.x, tid = threadIdx.x;
    const int u = u_idx[b], r = r_idx[b];
    __shared__ float hl[DIM], red0[DIM], red1[DIM], xv[DIM], sc[2];
    hl[tid] = ent[(size_t)u * DIM + tid];
    __syncthreads();
    const float* M = RotH + (size_t)r * DIM * DIM;
    float x = 0.f;
    for (int k = 0; k < DIM; ++k) x += hl[k] * M[k * DIM + tid];
    const float off = offH[r * DIM + tid];
    xv[tid] = x;  red0[tid] = x * x;  red1[tid] = x * off;
    __syncthreads();
    if (tid == 0) {
        float x2 = 0.f, xy = 0.f;
        for (int k = 0; k < DIM; ++k) { x2 += red0[k]; xy += red1[k]; }
        sc[0] = x2; sc[1] = xy;
    }
    __syncthreads();
    const float x2 = sc[0], xy = sc[1], y2 = y2H[r];
    const float den = 1.f + 2.f * xy + x2 * y2 + 1e-15f;
    TH[(size_t)b * DIM + tid] = ((1.f + 2.f * xy + y2) * xv[tid] + (1.f - x2) * off) / den;
}

// ---------------------------------------------------------------------------
// Kernel C: main scoring. grid = (B, N/64), 128 threads (4 waves).
// Stages X (64 tails x 64 dims) and Rot (64x64) in LDS, runs the rotation as
// fp32 WMMA (v_wmma_f32_16x16x4_f32), then Möbius add + Lorentz distance.
// ---------------------------------------------------------------------------
__global__ __launch_bounds__(128)
void score_kernel(const float* __restrict__ ent, const int* __restrict__ v_idx,
                  const int* __restrict__ u_idx, const int* __restrict__ r_idx,
                  const float* __restrict__ bias_h, const float* __restrict__ bias_t,
                  const float* __restrict__ RotT, const float* __restrict__ offT,
                  const float* __restrict__ y2T, const float* __restrict__ TH,
                  float* __restrict__ out, int N)
{
    const int b   = blockIdx.x;
    const int nt  = blockIdx.y;          // 64-tail tile index
    const int tid = threadIdx.x;
    const int r   = r_idx[b];

    __shared__ float Ml[DIM * 65];       // rotation, stride 65
    __shared__ float Xl[64 * 65];        // gathered tails
    __shared__ float Yl[64 * 65];        // rotated tails
    __shared__ float offl[DIM], thl[DIM];
    __shared__ int   vl[64];

    const float* M = RotT + (size_t)r * DIM * DIM;
    for (int idx = tid; idx < DIM * DIM; idx += 128) {
        int i = idx >> 6, j = idx & 63;
        Ml[i * 65 + j] = M[idx];
    }
    if (tid < DIM) { offl[tid] = offT[r * DIM + tid]; thl[tid] = TH[(size_t)b * DIM + tid]; }
    if (tid < 64)  { vl[tid] = v_idx[(size_t)b * N + nt * 64 + tid]; }
    __syncthreads();
    for (int idx = tid; idx < 64 * DIM; idx += 128) {
        int n = idx >> 6, k = idx & 63;
        Xl[n * 65 + k] = ent[(size_t)vl[n] * DIM + k];
    }
    __syncthreads();

    // Y(64x64) = X(64x64) @ M(64x64), fp32 WMMA, wave w owns rows 16w..16w+15.
    // A 16x4 layout: lanes 0-15 K={0,1}, lanes 16-31 K={2,3} (VGPR0/1).
    // B 4x16 / D 16x16: rows striped across lanes within a VGPR.
    const int wave  = tid >> 5;
    const int lane  = tid & 31;
    const int half  = lane >> 4;                 // 0 or 1
    const int mrow  = wave * 16 + (lane & 15);
    const int ncol0 = lane & 15;
    for (int ntile = 0; ntile < 4; ++ntile) {
        v8f acc = {0.f, 0.f, 0.f, 0.f, 0.f, 0.f, 0.f, 0.f};
        const int n = ntile * 16 + ncol0;
        #pragma unroll
        for (int ks = 0; ks < 16; ++ks) {
            const int kb = ks * 4 + half * 2;
            v2f a, bm;
            a.x  = Xl[mrow * 65 + kb];
            a.y  = Xl[mrow * 65 + kb + 1];
            bm.x = Ml[kb * 65 + n];
            bm.y = Ml[(kb + 1) * 65 + n];
            acc = __builtin_amdgcn_wmma_f32_16x16x4_f32(
                false, a, false, bm, (short)0, acc, false, false);
        }
        #pragma unroll
        for (int i = 0; i < 8; ++i)
            Yl[(wave * 16 + i + half * 8) * 65 + ntile * 16 + ncol0] = acc[i];
    }
    __syncthreads();

    // Möbius add + Lorentz score, one thread per tail.
    if (tid < 64) {
        const int n = tid;
        float x2 = 0.f, xy = 0.f;
        for (int j = 0; j < DIM; ++j) {
            float x = Yl[n * 65 + j];
            x2 += x * x; xy += x * offl[j];
        }
        const float y2  = y2T[r];
        const float den = 1.f + 2.f * xy + x2 * y2 + 1e-15f;
        const float ca  = 1.f + 2.f * xy + y2;
        const float cb  = 1.f - x2;
        float acc = 0.f, d0 = 0.f;
        for (int j = 0; j < DIM; ++j) {
            float ttj = (ca * Yl[n * 65 + j] + cb * offl[j]) / den;
            float dd  = ttj - thl[j];
            acc += dd * dd;
            if (j == 0) d0 = dd;
        }
        const float mkv = acc - 2.f * d0 * d0;
        out[(size_t)b * N + nt * 64 + n] =
            MARGIN_C - mkv + tanhf(bias_h[u_idx[b]]) + tanhf(bias_t[vl[n]]);
    }
}

// ---------------------------------------------------------------------------
extern "C" void kernel_launch(void* const* d_in, const int* in_sizes, int n_in,
                              void* d_out, int out_size, void* d_ws, size_t ws_size,
                              hipStream_t stream) {
    const float* ent  = (const float*)d_in[0];
    const float* Wfh  = (const float*)d_in[1];
    const float* Wft  = (const float*)d_in[2];
    const float* cenh = (const float*)d_in[3];
    const float* dirh = (const float*)d_in[4];
    const float* cent = (const float*)d_in[5];
    const float* dirt = (const float*)d_in[6];
    const float* bh   = (const float*)d_in[7];
    const float* bt   = (const float*)d_in[8];
    const float* fh   = (const float*)d_in[9];
    const float* ft   = (const float*)d_in[10];
    const int* u_idx  = (const int*)d_in[11];
    const int* r_idx  = (const int*)d_in[12];
    const int* v_idx  = (const int*)d_in[13];

    const int NR = in_sizes[3] / DIM;      // 237
    const int B  = in_sizes[11];           // 1024
    const int N  = in_sizes[13] / B;       // 512

    // workspace layout (~8.3 MB of fp32)
    float* ws   = (float*)d_ws;
    float* RotH = ws;
    float* RotT = RotH + (size_t)NR * DIM * DIM;
    float* offH = RotT + (size_t)NR * DIM * DIM;
    float* offT = offH + (size_t)NR * DIM;
    float* y2H  = offT + (size_t)NR * DIM;
    float* y2T  = y2H + NR;
    float* TH   = y2T + NR;

    rel_precompute_kernel<<<dim3(NR, 2), 64, 0, stream>>>(
        Wfh, Wft, cenh, dirh, cent, dirt, fh, ft,
        RotH, RotT, offH, offT, y2H, y2T);

    head_kernel<<<B, 64, 0, stream>>>(ent, u_idx, r_idx, RotH, offH, y2H, TH);

    score_kernel<<<dim3(B, N / 64), 128, 0, stream>>>(
        ent, v_idx, u_idx, r_idx, bh, bt, RotT, offT, y2T, TH,
        (float*)d_out, N);
}